// GINModel_38001870635069
// MI455X (gfx1250) — compile-verified
//
#include <hip/hip_runtime.h>
#include <cstdint>

typedef __attribute__((ext_vector_type(2))) float v2f;
typedef __attribute__((ext_vector_type(8))) float v8f;
typedef __attribute__((ext_vector_type(4))) unsigned int v4u;
typedef __attribute__((ext_vector_type(8))) int v8i;
typedef __attribute__((ext_vector_type(4))) int v4i;

#define LDSPAD 132   // 132 mod 64 = 4 -> 16 rows hit 16 distinct banks for A-frag reads
#define TILE_M 64
#define BN_EPS 1e-5f

// ---------------------------------------------------------------------------
// TDM: DMA a contiguous 128x128 fp32 row-major tile from global into LDS.
// D# packing per CDNA5 ISA sec 8.3/8.4 (2-D tensor, groups 2/3 unused -> zeros).
// Issue from ONE wave only (TDM ignores EXEC; one issuing wave = one DMA).
// This toolchain exposes the 6-arg builtin: (g0, g1, g2, g3, g_extra, cpol).
// ---------------------------------------------------------------------------
__device__ static inline void tdm_load_128x128_f32(const float* gsrc, float* lds_dst) {
    unsigned long long ga = (unsigned long long)(uintptr_t)gsrc;
    unsigned int ldsoff = (unsigned int)(uintptr_t)lds_dst;  // LDS byte offset = addr[31:0]
    v4u g0;
    g0.x = 0x1u;                                              // count=1, user mode
    g0.y = ldsoff;                                            // lds_addr  [63:32]
    g0.z = (unsigned int)(ga & 0xffffffffu);                  // global_addr lo [95:64]
    g0.w = (unsigned int)((ga >> 32) & 0x01ffffffu)           // global_addr hi [120:96]
           | 0x80000000u;                                     // type=2 ("image") [127:126]
    v8i g1;
    g1[0] = 0x00020000;            // workgroup_mask=0, data_size=2 (4 bytes)
    g1[1] = (int)(128u << 16);     // tensor_dim0[15:0]  @ bits[63:48]
    g1[2] = (int)(128u << 16);     // tensor_dim0 hi=0, tensor_dim1[15:0] @ [95:80]
    g1[3] = (int)(128u << 16);     // tensor_dim1 hi=0, tile_dim0=128 @ [127:112]
    g1[4] = 128;                   // tile_dim1=128 @ [143:128]
    g1[5] = 128;                   // tensor_dim0_stride lo32 = 128 elements
    g1[6] = 0;                     // stride hi, tensor_dim1_stride lo (unused, 2-D)
    g1[7] = 0;
    v4i z4 = {0, 0, 0, 0};
    v8i z8 = {0, 0, 0, 0, 0, 0, 0, 0};
    __builtin_amdgcn_tensor_load_to_lds(g0, g1, z4, z4, z8, 0);
}

// ---------------------------------------------------------------------------
// agg = src  (the "+x[i]" self term of GIN; scatter then adds neighbors)
// ---------------------------------------------------------------------------
__global__ void gin_copy_kernel(const float4* __restrict__ in,
                                float4* __restrict__ out, int n4) {
    int i = blockIdx.x * blockDim.x + threadIdx.x;
    int stride = gridDim.x * blockDim.x;
    for (; i < n4; i += stride) out[i] = in[i];
}

// ---------------------------------------------------------------------------
// agg[dst] += x[src] : one wave per edge, lane l owns float4 chunk l.
// agg (51 MB) is L2-resident on MI455X (192 MB L2) so the f32 atomics ping in L2.
// Next edge's source row is prefetched to hide the dependent-gather latency.
// ---------------------------------------------------------------------------
__global__ __launch_bounds__(256) void gin_scatter_kernel(
    const float* __restrict__ x, const int* __restrict__ ei,
    float* __restrict__ agg, int nE) {
    int lane = threadIdx.x & 31;
    int wid  = (blockIdx.x * (blockDim.x >> 5)) + (threadIdx.x >> 5);
    int nwaves = gridDim.x * (blockDim.x >> 5);
    for (int e = wid; e < nE; e += nwaves) {
        int e2 = e + nwaves;
        if (e2 < nE) {
            int s2 = ei[e2];
            __builtin_prefetch(x + (size_t)s2 * 128 + lane * 4, 0, 3);
        }
        int s = ei[e];
        int d = ei[nE + e];
        float4 v = ((const float4*)(x + (size_t)s * 128))[lane];
        float* dp = agg + (size_t)d * 128 + lane * 4;
        atomicAdd(dp + 0, v.x);
        atomicAdd(dp + 1, v.y);
        atomicAdd(dp + 2, v.z);
        atomicAdd(dp + 3, v.w);
    }
}

// ---------------------------------------------------------------------------
// Fused GIN MLP: out = relu(bn(relu(h@Wa+ba)@Wb+bb))
// 64 rows/block, 256 threads (8 waves), fp32 WMMA 16x16x4, K=128 in 32 steps.
// Both weight matrices are DMA'd into LDS by the TDM while waves stage the
// activation tile; ~161 KB LDS (within the 320 KB WGP budget).
// ---------------------------------------------------------------------------
__global__ __launch_bounds__(256) void gin_mlp_kernel(
    const float* __restrict__ hin,
    const float* __restrict__ Wa, const float* __restrict__ ba,
    const float* __restrict__ Wb, const float* __restrict__ bb,
    const float* __restrict__ bng, const float* __restrict__ bnb,
    const float* __restrict__ bnm, const float* __restrict__ bnv,
    float* __restrict__ out, int n_rows) {
    __shared__ float sWa[128 * 128];         // 64 KB (TDM destination)
    __shared__ float sWb[128 * 128];         // 64 KB (TDM destination)
    __shared__ float sH[TILE_M * LDSPAD];    // ~33 KB padded activation tile

    const int tid  = threadIdx.x;
    const int row0 = blockIdx.x * TILE_M;
    const int lane = tid & 31;
    const int wid  = tid >> 5;
    const int half = lane >> 4;       // 0: K pair {0,1}, 1: K pair {2,3}
    const int r16  = lane & 15;
    const int mr   = wid >> 1;        // row tile 0..3
    const int nc0  = (wid & 1) * 4;   // first of 4 col tiles

    // kick off both weight DMAs from wave 0 (TENSORcnt-tracked, in order)
    if (wid == 0) {
        tdm_load_128x128_f32(Wa, sWa);
        tdm_load_128x128_f32(Wb, sWb);
    }
    // stage h tile (coalesced float4, zero-pad tail rows) -- overlaps the DMA
    for (int i = tid; i < TILE_M * 32; i += 256) {
        int r = i >> 5, c4 = i & 31;
        int gr = row0 + r;
        float4 v = make_float4(0.f, 0.f, 0.f, 0.f);
        if (gr < n_rows) v = ((const float4*)(hin + (size_t)gr * 128))[c4];
        ((float4*)&sH[r * LDSPAD])[c4] = v;
    }
    if (wid == 0) __builtin_amdgcn_s_wait_tensorcnt(0);
    __syncthreads();

    v8f acc[4];
    // ---------------- GEMM 1: sH @ Wa ----------------
    #pragma unroll
    for (int t = 0; t < 4; t++) acc[t] = (v8f){0.f,0.f,0.f,0.f,0.f,0.f,0.f,0.f};
    for (int k0 = 0; k0 < 128; k0 += 4) {
        const float* ap = &sH[(mr * 16 + r16) * LDSPAD + k0 + 2 * half];
        v2f a; a.x = ap[0]; a.y = ap[1];
        #pragma unroll
        for (int t = 0; t < 4; t++) {
            int nc = (nc0 + t) * 16;
            v2f b;
            b.x = sWa[(k0 + 2 * half)     * 128 + nc + r16];
            b.y = sWa[(k0 + 2 * half + 1) * 128 + nc + r16];
            acc[t] = __builtin_amdgcn_wmma_f32_16x16x4_f32(
                false, a, false, b, (short)0, acc[t], false, false);
        }
    }
    __syncthreads();   // everyone done reading sH

    // epilogue 1: bias + relu -> sH (becomes A of GEMM 2)
    #pragma unroll
    for (int t = 0; t < 4; t++) {
        int col = (nc0 + t) * 16 + r16;
        float bias = ba[col];
        #pragma unroll
        for (int v = 0; v < 8; v++) {
            int row = mr * 16 + v + 8 * half;
            float val = acc[t][v] + bias;
            sH[row * LDSPAD + col] = val > 0.f ? val : 0.f;
        }
    }
    __syncthreads();

    // ---------------- GEMM 2: relu(...) @ Wb ----------------
    #pragma unroll
    for (int t = 0; t < 4; t++) acc[t] = (v8f){0.f,0.f,0.f,0.f,0.f,0.f,0.f,0.f};
    for (int k0 = 0; k0 < 128; k0 += 4) {
        const float* ap = &sH[(mr * 16 + r16) * LDSPAD + k0 + 2 * half];
        v2f a; a.x = ap[0]; a.y = ap[1];
        #pragma unroll
        for (int t = 0; t < 4; t++) {
            int nc = (nc0 + t) * 16;
            v2f b;
            b.x = sWb[(k0 + 2 * half)     * 128 + nc + r16];
            b.y = sWb[(k0 + 2 * half + 1) * 128 + nc + r16];
            acc[t] = __builtin_amdgcn_wmma_f32_16x16x4_f32(
                false, a, false, b, (short)0, acc[t], false, false);
        }
    }
    __syncthreads();   // done reading sH

    // epilogue 2: bias + BN + relu -> sH, then coalesced store
    #pragma unroll
    for (int t = 0; t < 4; t++) {
        int col = (nc0 + t) * 16 + r16;
        float bias  = bb[col];
        float scale = bng[col] * rsqrtf(bnv[col] + BN_EPS);
        float shift = bnb[col] - bnm[col] * scale;
        #pragma unroll
        for (int v = 0; v < 8; v++) {
            int row = mr * 16 + v + 8 * half;
            float val = (acc[t][v] + bias) * scale + shift;
            sH[row * LDSPAD + col] = val > 0.f ? val : 0.f;
        }
    }
    __syncthreads();
    for (int i = tid; i < TILE_M * 32; i += 256) {
        int r = i >> 5, c4 = i & 31;
        int gr = row0 + r;
        if (gr < n_rows)
            ((float4*)(out + (size_t)gr * 128))[c4] = ((const float4*)&sH[r * LDSPAD])[c4];
    }
}

// ---------------------------------------------------------------------------
// segmented mean-pool: batch is sorted, so accumulate in registers and flush
// an atomic only when the graph id changes within the 64-node chunk.
// ---------------------------------------------------------------------------
__global__ __launch_bounds__(128) void gin_pool_kernel(
    const float* __restrict__ h, const int* __restrict__ batch,
    float* __restrict__ pool, float* __restrict__ counts, int n_rows) {
    int base = blockIdx.x * 64;
    int col  = (threadIdx.x >> 5) * 32 + (threadIdx.x & 31);
    float acc = 0.f;
    int cur = -1;
    for (int i = 0; i < 64; i++) {
        int node = base + i;
        if (node >= n_rows) break;
        int g = batch[node];
        if (g != cur) {
            if (cur >= 0) atomicAdd(&pool[cur * 128 + col], acc);
            cur = g; acc = 0.f;
        }
        acc += h[(size_t)node * 128 + col];
    }
    if (cur >= 0) atomicAdd(&pool[cur * 128 + col], acc);

    if (threadIdx.x == 0) {
        int c2 = -1; float cnt = 0.f;
        for (int i = 0; i < 64; i++) {
            int node = base + i;
            if (node >= n_rows) break;
            int g = batch[node];
            if (g != c2) {
                if (c2 >= 0) atomicAdd(&counts[c2], cnt);
                c2 = g; cnt = 0.f;
            }
            cnt += 1.f;
        }
        if (c2 >= 0) atomicAdd(&counts[c2], cnt);
    }
}

__global__ void gin_zero_kernel(float* p, int n) {
    int i = blockIdx.x * blockDim.x + threadIdx.x;
    if (i < n) p[i] = 0.f;
}

__global__ void gin_finalize_kernel(const float* __restrict__ pool,
                                    const float* __restrict__ counts,
                                    float* __restrict__ out, int n) {
    int i = blockIdx.x * blockDim.x + threadIdx.x;
    if (i < n) out[i] = pool[i] / fmaxf(counts[i >> 7], 1.0f);
}

// ---------------------------------------------------------------------------
extern "C" void kernel_launch(void* const* d_in, const int* in_sizes, int n_in,
                              void* d_out, int out_size, void* d_ws, size_t ws_size,
                              hipStream_t stream) {
    const float* x     = (const float*)d_in[0];
    const int*   ei    = (const int*)d_in[1];    // per harness: integer -> const int*
    const int*   batch = (const int*)d_in[2];
    const float* w11 = (const float*)d_in[3];  const float* b11 = (const float*)d_in[4];
    const float* w12 = (const float*)d_in[5];  const float* b12 = (const float*)d_in[6];
    const float* w21 = (const float*)d_in[7];  const float* b21 = (const float*)d_in[8];
    const float* w22 = (const float*)d_in[9];  const float* b22 = (const float*)d_in[10];
    const float* w31 = (const float*)d_in[11]; const float* b31 = (const float*)d_in[12];
    const float* w32 = (const float*)d_in[13]; const float* b32 = (const float*)d_in[14];
    const float* bn1g = (const float*)d_in[15]; const float* bn1b = (const float*)d_in[16];
    const float* bn1m = (const float*)d_in[17]; const float* bn1v = (const float*)d_in[18];
    const float* bn2g = (const float*)d_in[19]; const float* bn2b = (const float*)d_in[20];
    const float* bn2m = (const float*)d_in[21]; const float* bn2v = (const float*)d_in[22];
    const float* bn3g = (const float*)d_in[23]; const float* bn3b = (const float*)d_in[24];
    const float* bn3m = (const float*)d_in[25]; const float* bn3v = (const float*)d_in[26];

    const int N = in_sizes[0] / 128;
    const int E = in_sizes[1] / 2;
    const int GO = out_size;              // G * 128

    float* bufA   = (float*)d_ws;
    float* bufB   = bufA + (size_t)N * 128;
    float* pool   = bufB + (size_t)N * 128;
    float* counts = pool + GO;

    const int n4 = N * 32;                // float4 count of an activation buffer
    const int mlpBlocks = (N + TILE_M - 1) / TILE_M;
    float* outp = (float*)d_out;

    // ----- layer 1 -----
    gin_copy_kernel<<<2048, 256, 0, stream>>>((const float4*)x, (float4*)bufA, n4);
    gin_scatter_kernel<<<4096, 256, 0, stream>>>(x, ei, bufA, E);
    gin_mlp_kernel<<<mlpBlocks, 256, 0, stream>>>(bufA, w11, b11, w12, b12,
                                                  bn1g, bn1b, bn1m, bn1v, bufB, N);
    // ----- layer 2 (MLP is block-local => in-place safe) -----
    gin_copy_kernel<<<2048, 256, 0, stream>>>((const float4*)bufB, (float4*)bufA, n4);
    gin_scatter_kernel<<<4096, 256, 0, stream>>>(bufB, ei, bufA, E);
    gin_mlp_kernel<<<mlpBlocks, 256, 0, stream>>>(bufA, w21, b21, w22, b22,
                                                  bn2g, bn2b, bn2m, bn2v, bufA, N);
    // ----- layer 3 -----
    gin_copy_kernel<<<2048, 256, 0, stream>>>((const float4*)bufA, (float4*)bufB, n4);
    gin_scatter_kernel<<<4096, 256, 0, stream>>>(bufA, ei, bufB, E);
    gin_mlp_kernel<<<mlpBlocks, 256, 0, stream>>>(bufB, w31, b31, w32, b32,
                                                  bn3g, bn3b, bn3m, bn3v, bufB, N);
    // ----- global mean pool -----
    gin_zero_kernel<<<(GO + GO / 128 + 255) / 256, 256, 0, stream>>>(pool, GO + GO / 128);
    gin_pool_kernel<<<(N + 63) / 64, 128, 0, stream>>>(bufB, batch, pool, counts, N);
    gin_finalize_kernel<<<(GO + 255) / 256, 256, 0, stream>>>(pool, counts, outp, GO);
}